// TemporalExtensionShift_25769804037
// MI455X (gfx1250) — compile-verified
//
#include <hip/hip_runtime.h>

// Temporal Shift Module gather: y[n,c,t,:,:] = x[n,c,t+s(c),:,:] with
//   s(c) = +3 for c in [0,32), -3 for c in [32,64), 0 otherwise,
// zero-filled where t+s(c) is out of [0,16).
//
// Pure streaming data movement: 205.5 MB in + 205.5 MB out, 0 FLOPs.
// Roofline at 23.3 TB/s HBM: ~17.6 us. Strategy: 128-bit coalesced
// loads/stores with non-temporal (TH=NT) cache hints, since the 411 MB
// working set exceeds the 192 MB L2 and has zero reuse.

typedef float v4f __attribute__((ext_vector_type(4)));

namespace {
constexpr unsigned kN   = 16;
constexpr unsigned kC   = 256;
constexpr unsigned kT   = 16;          // power of 2 -> mask
constexpr unsigned kHW  = 28 * 28;     // 784 floats per t-plane
constexpr unsigned kHW4 = kHW / 4;     // 196 float4 per t-plane
constexpr unsigned kFold = kC / 8;     // 32
constexpr unsigned kTotal4 = kN * kC * kT * kHW4;  // 12,845,056 float4
constexpr unsigned kBlock  = 256;                  // 8 wave32 waves
constexpr unsigned kGrid   = kTotal4 / kBlock;     // 50,176 (exact)
static_assert(kTotal4 % kBlock == 0, "exact grid");
}  // namespace

__global__ __launch_bounds__(kBlock)
void tsm_shift_kernel(const float* __restrict__ x, float* __restrict__ y) {
    // Flat float4 index over the whole tensor; consecutive lanes ->
    // consecutive 16B chunks -> perfectly coalesced b128 accesses.
    const unsigned i = blockIdx.x * kBlock + threadIdx.x;

    // plane = n*C*T + c*T + t  (one division by 196; T,C are pow2 masks)
    const unsigned plane = i / kHW4;
    const unsigned t = plane & (kT - 1);
    const unsigned c = (plane >> 4) & (kC - 1);

    // Temporal shift for this channel block.
    const int s = (c < kFold) ? 3 : ((c < 2 * kFold) ? -3 : 0);
    const int ts = (int)t + s;

    v4f val = {0.f, 0.f, 0.f, 0.f};
    if ((unsigned)ts < kT) {
        // Shifting by s t-planes == shifting the flat float4 index by s*196.
        const v4f* src = (const v4f*)x + (int)i + s * (int)kHW4;
        val = __builtin_nontemporal_load(src);   // global_load_b128, TH=NT
    }
    __builtin_nontemporal_store(val, (v4f*)y + i);  // global_store_b128, TH=NT
}

extern "C" void kernel_launch(void* const* d_in, const int* in_sizes, int n_in,
                              void* d_out, int out_size, void* d_ws, size_t ws_size,
                              hipStream_t stream) {
    (void)in_sizes; (void)n_in; (void)d_ws; (void)ws_size; (void)out_size;
    const float* x = (const float*)d_in[0];
    // d_in[1] is the one-hot depthwise weight; its semantics (shift +3/-3/0
    // per channel-fold) are baked into the kernel, matching the reference's
    // fixed __init__-time construction.
    float* y = (float*)d_out;

    tsm_shift_kernel<<<dim3(kGrid), dim3(kBlock), 0, stream>>>(x, y);
}